// HierarchicalMultinomialRegression_32212254720367
// MI455X (gfx1250) — compile-verified
//
#include <hip/hip_runtime.h>
#include <hip/hip_bf16.h>
#include <math.h>

typedef __attribute__((ext_vector_type(2))) float v2f;
typedef __attribute__((ext_vector_type(8))) float v8f;

#define KM1 7
#define TT  20
#define NF  64

// Precompute tiny constants into d_ws:
//  ws[0..48]   : rho[l][k] = tanh(raw_rho[l][k])
//  ws[49..97]  : sd[l][k]  = 1/sqrt(1-rho^2) * raw_chol[k][l][l]
//  ws[98..146] : M[k][j]   = sum_i raw_chol[k][i][j]
__global__ void hmr_prep_kernel(const float* __restrict__ raw_rho,
                                const float* __restrict__ raw_chol,
                                float* __restrict__ ws) {
  int i = threadIdx.x;
  if (i < 49) {
    int l = i / KM1, k = i % KM1;
    float r = tanhf(raw_rho[l * KM1 + k]);
    ws[i] = r;
    float sc = rsqrtf(1.0f - r * r);
    ws[49 + i] = sc * raw_chol[k * 49 + l * KM1 + l];
    // reuse thread index as (k,j) for M
    int kk = l, j = k;
    float m = 0.f;
    #pragma unroll
    for (int ii = 0; ii < KM1; ++ii) m += raw_chol[kk * 49 + ii * KM1 + j];
    ws[98 + i] = m;
  }
}

__global__ void __launch_bounds__(256)
hmr_main_kernel(const float* __restrict__ X,
                const float* __restrict__ beta,
                const float* __restrict__ eps,
                const int* __restrict__ batter,
                const int* __restrict__ league,
                const int* __restrict__ season,
                const float* __restrict__ ws,
                float* __restrict__ out,
                int n) {
  __shared__ float s_fixed[256][9];   // padded stride: no LDS bank conflicts
  __shared__ float s_c[147];

  int tid = threadIdx.x;
  if (tid < 147) s_c[tid] = ws[tid];

  int wave = tid >> 5;
  int lane = tid & 31;
  int wave_base = blockIdx.x * 256 + wave * 32;

  int col  = lane & 15;              // N-column of tile / M-row of A frag
  int koff = (lane >> 4) * 2;        // K sub-offset per ISA A/B layout

  // ------------- Phase 1: fixed_effects = X @ beta --------------------------
  if (wave_base + 32 <= n) {
    // WMMA fast path: every row in this wave is valid -> zero divergence,
    // EXEC is all-ones around the WMMAs, all loads unconditional.
    int colc = (col < KM1) ? col : (KM1 - 1);   // clamped (in-bounds) address
    float bmask = (col < KM1) ? 1.0f : 0.0f;    // zero value, not the load

    v2f bfrag[16];
    #pragma unroll
    for (int i = 0; i < 16; ++i) {
      int k = i * 4 + koff;
      v2f b;
      b.x = beta[k * KM1 + colc] * bmask;
      b.y = beta[(k + 1) * KM1 + colc] * bmask;
      bfrag[i] = b;
    }

    #pragma unroll
    for (int h = 0; h < 2; ++h) {
      const float* xr = X + (size_t)(wave_base + h * 16 + col) * NF + koff;
      // batch all A fragments as unconditional 8-byte loads
      v2f afrag[16];
      #pragma unroll
      for (int i = 0; i < 16; ++i)
        afrag[i] = *(const v2f*)(xr + i * 4);

      v8f acc = {};
      #pragma unroll
      for (int i = 0; i < 16; ++i)
        acc = __builtin_amdgcn_wmma_f32_16x16x4_f32(
            false, afrag[i], false, bfrag[i], (short)0, acc, false, false);

      // D layout: VGPR r, lanes 0-15 -> M=r ; lanes 16-31 -> M=8+r ; N=lane&15
      if (col < KM1) {
        #pragma unroll
        for (int r = 0; r < 8; ++r) {
          int m = r + 8 * (lane >> 4);
          s_fixed[wave * 32 + h * 16 + m][col] = acc[r];
        }
      }
    }
  } else if (wave_base < n) {
    // ragged tail (never taken for n % 32 == 0): plain VALU dot products
    int row = wave_base + lane;
    if (row < n) {
      const float* xr = X + (size_t)row * NF;
      #pragma unroll
      for (int j = 0; j < KM1; ++j) {
        float acc = 0.f;
        for (int k = 0; k < NF; ++k)
          acc = fmaf(xr[k], beta[k * KM1 + j], acc);
        s_fixed[wave * 32 + lane][j] = acc;
      }
    }
  }
  __syncthreads();

  // ------------- Phase 2: lazy AR(1) scan per output row --------------------
  int ni = wave_base + lane;
  if (ni < n) {
    int b = batter[ni], l = league[ni], t = season[ni];
    const float* erow = eps + ((size_t)(b * KM1 + l) * TT) * KM1;
    __builtin_prefetch(erow, 0, 0);

    float rho_l[KM1], sd_l[KM1], Mr[KM1][KM1];
    #pragma unroll
    for (int k = 0; k < KM1; ++k) {
      rho_l[k] = s_c[l * KM1 + k];
      sd_l[k]  = s_c[49 + l * KM1 + k];
    }
    #pragma unroll
    for (int k = 0; k < KM1; ++k)
      #pragma unroll
      for (int j = 0; j < KM1; ++j)
        Mr[k][j] = s_c[98 + k * KM1 + j];

    float e[KM1], u[KM1];
    #pragma unroll
    for (int k = 0; k < KM1; ++k) e[k] = erow[k];
    #pragma unroll
    for (int k = 0; k < KM1; ++k) u[k] = sd_l[k] * e[k];

    for (int s = 1; s <= t; ++s) {
      const float* ep = erow + s * KM1;
      #pragma unroll
      for (int k = 0; k < KM1; ++k) e[k] = ep[k];
      float nz[KM1];
      #pragma unroll
      for (int j = 0; j < KM1; ++j) nz[j] = 0.f;
      #pragma unroll
      for (int k = 0; k < KM1; ++k)
        #pragma unroll
        for (int j = 0; j < KM1; ++j)
          nz[j] = fmaf(e[k], Mr[k][j], nz[j]);
      #pragma unroll
      for (int j = 0; j < KM1; ++j) u[j] = fmaf(rho_l[j], u[j], nz[j]);
    }

    // outputs: logits (n,8) then eps_sample (n,7), concatenated flat
    float* lp = out + (size_t)ni * 8;
    lp[0] = 0.f;
    #pragma unroll
    for (int k = 0; k < KM1; ++k)
      lp[1 + k] = s_fixed[wave * 32 + lane][k] + u[k];

    float* sp = out + (size_t)n * 8 + (size_t)ni * KM1;
    #pragma unroll
    for (int k = 0; k < KM1; ++k) sp[k] = e[k];
  }
}

extern "C" void kernel_launch(void* const* d_in, const int* in_sizes, int n_in,
                              void* d_out, int out_size, void* d_ws, size_t ws_size,
                              hipStream_t stream) {
  const float* X        = (const float*)d_in[0];
  const float* beta     = (const float*)d_in[1];
  const float* raw_rho  = (const float*)d_in[2];
  const float* raw_chol = (const float*)d_in[3];
  const float* eps      = (const float*)d_in[4];
  const int*   batter   = (const int*)d_in[5];
  const int*   league   = (const int*)d_in[6];
  const int*   season   = (const int*)d_in[7];
  float* out = (float*)d_out;
  float* ws  = (float*)d_ws;

  int n = in_sizes[0] / NF;

  hmr_prep_kernel<<<1, 64, 0, stream>>>(raw_rho, raw_chol, ws);
  int grid = (n + 255) / 256;
  hmr_main_kernel<<<grid, 256, 0, stream>>>(X, beta, eps, batter, league, season,
                                            ws, out, n);
}